// GATNodeEncoder_71871982731304
// MI455X (gfx1250) — compile-verified
//
#include <hip/hip_runtime.h>
#include <math.h>

#define IN_CH 128
#define HID 32
#define HEADS1 4
#define OUT_CH 64
#define NEG_SLOPE 0.2f

typedef __attribute__((ext_vector_type(2))) float v2f;
typedef __attribute__((ext_vector_type(8))) float v8f;

// ---------------- utility ----------------

__global__ void k_fill_i32(int* __restrict__ p, int val, long n) {
    long i = blockIdx.x * (long)blockDim.x + threadIdx.x;
    if (i < n) p[i] = val;
}

// W[K][ldn] -> Wt[ldn][K]
__global__ void k_transpose(const float* __restrict__ W, float* __restrict__ Wt,
                            int K, int ldn) {
    int i = blockIdx.x * blockDim.x + threadIdx.x;
    if (i < K * ldn) {
        int k = i / ldn, c = i % ldn;
        Wt[c * K + k] = W[i];
    }
}

__device__ __forceinline__ float leaky(float x) {
    return x > 0.0f ? x : x * NEG_SLOPE;
}

// order-preserving float <-> signed-int key (involution), for atomicMax on int
__device__ __forceinline__ int f2key(float f) {
    int b = __float_as_int(f);
    return b >= 0 ? b : (int)(0x80000000u - (unsigned)b);
}
__device__ __forceinline__ float key2f(int k) {
    int b = k >= 0 ? k : (int)(0x80000000u - (unsigned)k);
    return __int_as_float(b);
}

__device__ __forceinline__ void edge_nodes(const int* __restrict__ ei, long E, long eid,
                                           int& s, int& d) {
    if (eid < E) { s = ei[eid]; d = ei[E + eid]; }
    else         { s = d = (int)(eid - E); }  // self loop
}

// ---------------- fp32 WMMA GEMM: H[n][LDN] = X[n][K] @ W[K][LDN] ----------------
// Wt is W pre-transposed to [LDN][K] so each lane's B k-pair is contiguous.
// Block = 128 threads (4 waves). Wave tile: m0 = blockIdx.x*16,
// n0 = (blockIdx.y*4 + waveId)*16. V_WMMA_F32_16X16X4_F32, K fully unrolled.
template <int K_T, int LDN_T>
__global__ void k_gemm_wmma(const float* __restrict__ X, const float* __restrict__ Wt,
                            float* __restrict__ H, int n_nodes) {
    int lane = threadIdx.x & 31;
    int wave = threadIdx.x >> 5;
    int m0 = blockIdx.x * 16;
    int n0 = (blockIdx.y * 4 + wave) * 16;
    if (n0 >= LDN_T) return;

    int r  = lane & 15;        // A row / B col within tile
    int kh = (lane >> 4) * 2;  // k sub-offset for this half-wave

    // clamp A row: an A row only contributes to its own (unstored) output row,
    // so clamping keeps loads in-bounds without affecting valid outputs.
    int arow = m0 + r;
    if (arow >= n_nodes) arow = n_nodes - 1;
    int bcol = n0 + r;

    const float* ap = X  + (long)arow * K_T + kh;  // 8B aligned (K_T mult of 4)
    const float* bp = Wt + (long)bcol * K_T + kh;

    v8f acc = {};
#pragma unroll
    for (int k0 = 0; k0 < K_T; k0 += 4) {
        v2f a = *(const v2f*)(ap + k0);   // global_load_b64, imm offset
        v2f b = *(const v2f*)(bp + k0);   // global_load_b64, imm offset
        acc = __builtin_amdgcn_wmma_f32_16x16x4_f32(false, a, false, b,
                                                    (short)0, acc, false, false);
    }

    // C/D layout: VGPR i -> rows m0+i (lanes 0-15) and m0+8+i (lanes 16-31)
    int half = lane >> 4;
    int col  = n0 + (lane & 15);
#pragma unroll
    for (int i = 0; i < 8; i++) {
        int row = m0 + i + half * 8;
        if (row < n_nodes) H[(long)row * LDN_T + col] = acc[i];
    }
}

// ---------------- per-node attention logits ----------------
__global__ void k_att(const float* __restrict__ H, const float* __restrict__ atts,
                      const float* __restrict__ attd, float* __restrict__ as_,
                      float* __restrict__ ad_, int n, int heads, int ch) {
    long i = blockIdx.x * (long)blockDim.x + threadIdx.x;
    if (i >= (long)n * heads) return;
    int node = (int)(i / heads), h = (int)(i % heads);
    const float* hp = H + (long)node * heads * ch + (long)h * ch;
    float ss = 0.0f, dd = 0.0f;
    for (int c = 0; c < ch; c++) {
        float v = hp[c];
        ss += v * atts[h * ch + c];
        dd += v * attd[h * ch + c];
    }
    as_[i] = ss;
    ad_[i] = dd;
}

// ---------------- edge pass 1: segment max (as int keys) ----------------
template <int HEADS_T>
__global__ void k_edge_max(const int* __restrict__ ei, long E, int n,
                           const float* __restrict__ as_, const float* __restrict__ ad_,
                           int* __restrict__ mkey) {
    long eid = blockIdx.x * (long)blockDim.x + threadIdx.x;
    if (eid >= E + n) return;
    int s, d; edge_nodes(ei, E, eid, s, d);
#pragma unroll
    for (int h = 0; h < HEADS_T; h++) {
        float e = leaky(as_[(long)s * HEADS_T + h] + ad_[(long)d * HEADS_T + h]);
        atomicMax(&mkey[(long)d * HEADS_T + h], f2key(e));
    }
}

// ---------------- edge pass 2: softmax denominator ----------------
template <int HEADS_T>
__global__ void k_edge_denom(const int* __restrict__ ei, long E, int n,
                             const float* __restrict__ as_, const float* __restrict__ ad_,
                             const int* __restrict__ mkey, float* __restrict__ denom) {
    long eid = blockIdx.x * (long)blockDim.x + threadIdx.x;
    if (eid >= E + n) return;
    int s, d; edge_nodes(ei, E, eid, s, d);
#pragma unroll
    for (int h = 0; h < HEADS_T; h++) {
        float e  = leaky(as_[(long)s * HEADS_T + h] + ad_[(long)d * HEADS_T + h]);
        float m  = key2f(mkey[(long)d * HEADS_T + h]);
        float ex = expf(e - m);
        atomicAdd(&denom[(long)d * HEADS_T + h], ex);
    }
}

// ---------------- edge pass 3: weighted scatter-add (one wave32 per edge) ----------------
template <int HEADS_T, int CH_T>
__global__ void k_edge_aggr(const int* __restrict__ ei, long E, int n,
                            const float* __restrict__ Hf,
                            const float* __restrict__ as_, const float* __restrict__ ad_,
                            const int* __restrict__ mkey, const float* __restrict__ denom,
                            float* __restrict__ out) {
    long gid  = blockIdx.x * (long)blockDim.x + threadIdx.x;
    long eid  = gid >> 5;
    int  lane = (int)(gid & 31);
    if (eid >= E + n) return;
    int s, d; edge_nodes(ei, E, eid, s, d);

    // each lane computes alpha for head (lane & (HEADS_T-1)), then broadcast
    int myh = lane & (HEADS_T - 1);
    float e  = leaky(as_[(long)s * HEADS_T + myh] + ad_[(long)d * HEADS_T + myh]);
    float m  = key2f(mkey[(long)d * HEADS_T + myh]);
    float ex = expf(e - m);
    float myalpha = ex / (denom[(long)d * HEADS_T + myh] + 1e-16f);

    float alpha[HEADS_T];
#pragma unroll
    for (int h = 0; h < HEADS_T; h++) alpha[h] = __shfl(myalpha, h, 32);

    const int C = HEADS_T * CH_T;
#pragma unroll
    for (int cc = lane; cc < C; cc += 32) {
        int h = cc / CH_T;
        atomicAdd(&out[(long)d * C + cc], Hf[(long)s * C + cc] * alpha[h]);
    }
}

// ---------------- bias (+optional ReLU) ----------------
__global__ void k_bias_act(float* __restrict__ out, const float* __restrict__ b,
                           long n, int C, int do_relu) {
    long i = blockIdx.x * (long)blockDim.x + threadIdx.x;
    if (i >= n * C) return;
    float v = out[i] + b[i % C];
    out[i] = do_relu ? fmaxf(v, 0.0f) : v;
}

// ---------------- launch ----------------
extern "C" void kernel_launch(void* const* d_in, const int* in_sizes, int n_in,
                              void* d_out, int out_size, void* d_ws, size_t ws_size,
                              hipStream_t stream) {
    const float* x     = (const float*)d_in[0];
    const int*   ei    = (const int*)  d_in[1];
    const float* W1    = (const float*)d_in[2];
    const float* atts1 = (const float*)d_in[3];
    const float* attd1 = (const float*)d_in[4];
    const float* b1    = (const float*)d_in[5];
    const float* W2    = (const float*)d_in[6];
    const float* atts2 = (const float*)d_in[7];
    const float* attd2 = (const float*)d_in[8];
    const float* b2    = (const float*)d_in[9];

    const int  n  = in_sizes[0] / IN_CH;     // 50000
    const long E  = in_sizes[1] / 2;         // 800000
    const long ET = E + n;                   // with self loops

    // workspace carve-out
    char*  ws  = (char*)d_ws;
    size_t off = 0;
    auto carve = [&](size_t bytes) -> char* {
        char* p = ws + off;
        off += (bytes + 255) & ~(size_t)255;
        return p;
    };
    float* h1   = (float*)carve((size_t)n * 128 * 4);
    float* agg1 = (float*)carve((size_t)n * 128 * 4);   // becomes relu(out1+b1)
    float* as1  = (float*)carve((size_t)n * HEADS1 * 4);
    float* ad1  = (float*)carve((size_t)n * HEADS1 * 4);
    int*   m1   = (int*)  carve((size_t)n * HEADS1 * 4);
    float* den1 = (float*)carve((size_t)n * HEADS1 * 4);
    float* h2   = (float*)carve((size_t)n * OUT_CH * 4);
    float* as2  = (float*)carve((size_t)n * 4);
    float* ad2  = (float*)carve((size_t)n * 4);
    int*   m2   = (int*)  carve((size_t)n * 4);
    float* den2 = (float*)carve((size_t)n * 4);
    float* Wt1  = (float*)carve((size_t)IN_CH * 128 * 4);    // W1^T [128][128]
    float* Wt2  = (float*)carve((size_t)128 * OUT_CH * 4);   // W2^T [64][128]
    float* out  = (float*)d_out;

    const int BT = 256;
    auto blocks = [&](long t) { return (unsigned)((t + BT - 1) / BT); };
    const int NEG_INF_KEY = (int)0x80000000;

    // ---- init + weight transposes ----
    k_fill_i32<<<blocks((long)n * HEADS1), BT, 0, stream>>>(m1, NEG_INF_KEY, (long)n * HEADS1);
    k_fill_i32<<<blocks((long)n * HEADS1), BT, 0, stream>>>((int*)den1, 0, (long)n * HEADS1);
    k_fill_i32<<<blocks((long)n * 128), BT, 0, stream>>>((int*)agg1, 0, (long)n * 128);
    k_fill_i32<<<blocks((long)n), BT, 0, stream>>>(m2, NEG_INF_KEY, (long)n);
    k_fill_i32<<<blocks((long)n), BT, 0, stream>>>((int*)den2, 0, (long)n);
    k_fill_i32<<<blocks((long)n * OUT_CH), BT, 0, stream>>>((int*)out, 0, (long)n * OUT_CH);
    k_transpose<<<blocks(IN_CH * 128), BT, 0, stream>>>(W1, Wt1, IN_CH, 128);
    k_transpose<<<blocks(128 * OUT_CH), BT, 0, stream>>>(W2, Wt2, 128, OUT_CH);

    // ---- layer 1 ----
    {
        dim3 g((n + 15) / 16, (128 + 63) / 64);
        k_gemm_wmma<IN_CH, 128><<<g, 128, 0, stream>>>(x, Wt1, h1, n);
    }
    k_att<<<blocks((long)n * HEADS1), BT, 0, stream>>>(h1, atts1, attd1, as1, ad1, n, HEADS1, HID);
    k_edge_max<HEADS1><<<blocks(ET), BT, 0, stream>>>(ei, E, n, as1, ad1, m1);
    k_edge_denom<HEADS1><<<blocks(ET), BT, 0, stream>>>(ei, E, n, as1, ad1, m1, den1);
    k_edge_aggr<HEADS1, HID><<<blocks(ET * 32), BT, 0, stream>>>(ei, E, n, h1, as1, ad1, m1, den1, agg1);
    k_bias_act<<<blocks((long)n * 128), BT, 0, stream>>>(agg1, b1, n, 128, 1);

    // ---- layer 2 (heads=1, ch=64, mean over 1 head == identity) ----
    {
        dim3 g((n + 15) / 16, (OUT_CH + 63) / 64);
        k_gemm_wmma<128, OUT_CH><<<g, 128, 0, stream>>>(agg1, Wt2, h2, n);
    }
    k_att<<<blocks((long)n), BT, 0, stream>>>(h2, atts2, attd2, as2, ad2, n, 1, OUT_CH);
    k_edge_max<1><<<blocks(ET), BT, 0, stream>>>(ei, E, n, as2, ad2, m2);
    k_edge_denom<1><<<blocks(ET), BT, 0, stream>>>(ei, E, n, as2, ad2, m2, den2);
    k_edge_aggr<1, OUT_CH><<<blocks(ET * 32), BT, 0, stream>>>(ei, E, n, h2, as2, ad2, m2, den2, out);
    k_bias_act<<<blocks((long)n * OUT_CH), BT, 0, stream>>>(out, b2, n, OUT_CH, 0);
}